// ConvolutionalLayer_27479200760215
// MI455X (gfx1250) — compile-verified
//
#include <hip/hip_runtime.h>

typedef float v2f __attribute__((ext_vector_type(2)));
typedef float v8f __attribute__((ext_vector_type(8)));

#define C_IN   16
#define H_IMG  32
#define W_IMG  32
#define NCOUT  64
#define OWW    16
#define PIX    256   // 16x16 output positions per image
#define N0     36
#define N1     6

__device__ __forceinline__ float sigmoidf(float v) {
    return 1.0f / (1.0f + __expf(-v));
}

// Async global->LDS copy (CDNA5 GLOBAL_LOAD_ASYNC_TO_LDS_B32, GV addressing).
// LDS address = low 32 bits of the flat shared-memory pointer (aperture is in
// addr[63:32], low bits are the workgroup-relative LDS byte address).
__device__ __forceinline__ void async_b32(const void* gsrc, void* ldst) {
    unsigned      l = (unsigned)(uintptr_t)ldst;
    unsigned long long g = (unsigned long long)(uintptr_t)gsrc;
    asm volatile("global_load_async_to_lds_b32 %0, %1, off"
                 :: "v"(l), "v"(g) : "memory");
}
__device__ __forceinline__ void wait_async0() {
    asm volatile("s_wait_asynccnt 0x0" ::: "memory");
}

// Gather one patch element: j indexes (c, kh, kw) with c slowest; stride 2, pad 2.
__device__ __forceinline__ float gather_x(const float* __restrict__ x, int bImg,
                                          int j, int oh, int ow) {
    int c   = j / 25;
    int rem = j - 25 * c;
    int kh  = rem / 5;
    int kw  = rem - 5 * kh;
    int row = 2 * oh + kh - 2;
    int col = 2 * ow + kw - 2;
    bool ok = (row >= 0) && (row < H_IMG) && (col >= 0) && (col < W_IMG);
    int rc = ok ? row : 0;
    int cc = ok ? col : 0;
    float v = x[(((size_t)bImg * C_IN + c) * H_IMG + rc) * W_IMG + cc];
    return ok ? v : 0.0f;   // cndmask only, EXEC untouched
}

// Exact MSB-first multilinear contraction over a 64-entry LUT (matches reference).
__device__ __forceinline__ float lut64(const float* sw, const float g[6]) {
    float a[32];
#pragma unroll
    for (int r = 0; r < 32; ++r) a[r] = sw[r] * (1.0f - g[0]) + sw[r + 32] * g[0];
#pragma unroll
    for (int r = 0; r < 16; ++r) a[r] = a[r] * (1.0f - g[1]) + a[r + 16] * g[1];
#pragma unroll
    for (int r = 0; r < 8; ++r)  a[r] = a[r] * (1.0f - g[2]) + a[r + 8] * g[2];
#pragma unroll
    for (int r = 0; r < 4; ++r)  a[r] = a[r] * (1.0f - g[3]) + a[r + 4] * g[3];
    a[0] = a[0] * (1.0f - g[4]) + a[2] * g[4];
    a[1] = a[1] * (1.0f - g[4]) + a[3] * g[4];
    return a[0] * (1.0f - g[5]) + a[1] * g[5];
}

__global__ __launch_bounds__(512)
void lutconv_kernel(const float* __restrict__ x,
                    const int*   __restrict__ idx0, const float* __restrict__ table0,
                    const int*   __restrict__ idx1, const float* __restrict__ table1,
                    const int*   __restrict__ idx2, const float* __restrict__ table2,
                    float*       __restrict__ out) {
    __shared__ float ldsSW0[N0 * 64];   // sigmoid(table0[t])
    __shared__ float ldsSW1[N1 * 64];
    __shared__ float ldsSW2[64];
    __shared__ float ldsH0[N0 * PIX];   // layer-0 activations [node][pixel]
    __shared__ float ldsH1[N1 * PIX];

    const int t    = blockIdx.x;   // output channel
    const int bImg = blockIdx.y;   // image
    const int tid  = threadIdx.x;  // 0..511

    // ---- stage raw tables via async global->LDS DMA, then sigmoid in place ----
    for (int i = tid; i < N0 * 64; i += 512) async_b32(&table0[t * N0 * 64 + i], &ldsSW0[i]);
    for (int i = tid; i < N1 * 64; i += 512) async_b32(&table1[t * N1 * 64 + i], &ldsSW1[i]);
    if (tid < 64)                            async_b32(&table2[t * 64 + tid],    &ldsSW2[tid]);
    wait_async0();          // drain this wave's ASYNCcnt before signaling barrier
    __syncthreads();
    for (int i = tid; i < N0 * 64; i += 512) ldsSW0[i] = sigmoidf(ldsSW0[i]);
    for (int i = tid; i < N1 * 64; i += 512) ldsSW1[i] = sigmoidf(ldsSW1[i]);
    if (tid < 64)                            ldsSW2[tid] = sigmoidf(ldsSW2[tid]);
    __syncthreads();

    // ---- layer 0: 36 nodes, WMMA f32 16x16x4 does the inner (ahi x alo) GEMM ----
    // wave w handles output row oh=w; lane l (and l+16) handle pixel ow = l&15.
    const int  lane = tid & 31;
    const int  wv   = tid >> 5;        // 0..15 == oh
    const int  m    = lane & 15;       // A row (ahi) / B column (pixel)
    const bool hi   = lane >= 16;      // lane half carries K slots {2,3} vs {0,1}
    const int  oh   = wv;
    const int  ow   = m;
    const int  p    = oh * OWW + ow;
    const int  ks   = hi ? 2 : 0;
    const int  ib0  = t * N0 * 6;      // idx0 base (wave-uniform -> s_load)

    for (int n = 0; n < N0; ++n) {
        const int j0 = idx0[ib0 + n * 6 + 0], j1 = idx0[ib0 + n * 6 + 1];
        const int j2 = idx0[ib0 + n * 6 + 2], j3 = idx0[ib0 + n * 6 + 3];
        const int j4 = idx0[ib0 + n * 6 + 4], j5 = idx0[ib0 + n * 6 + 5];
        const float g0 = gather_x(x, bImg, j0, oh, ow);
        const float g1 = gather_x(x, bImg, j1, oh, ow);
        const float g2 = gather_x(x, bImg, j2, oh, ow);
        const float g3 = gather_x(x, bImg, j3, oh, ow);
        const float g4 = gather_x(x, bImg, j4, oh, ow);
        const float g5 = gather_x(x, bImg, j5, oh, ow);

        // corner products: P_hi over bits 0-2 (MSBs), P_lo over bits 3-5 (LSBs)
        const float f0 = 1.0f - g0, f1 = 1.0f - g1, f2 = 1.0f - g2;
        const float f3 = 1.0f - g3, f4 = 1.0f - g4, f5 = 1.0f - g5;
        const float h00 = f0 * f1, h01 = f0 * g1, h10 = g0 * f1, h11 = g0 * g1;
        const float Phi0 = h00 * f2, Phi1 = h00 * g2, Phi2 = h01 * f2, Phi3 = h01 * g2;
        const float Phi4 = h10 * f2, Phi5 = h10 * g2, Phi6 = h11 * f2, Phi7 = h11 * g2;
        const float l00 = f3 * f4, l01 = f3 * g4, l10 = g3 * f4, l11 = g3 * g4;
        const float Plo0 = l00 * f5, Plo1 = l00 * g5, Plo2 = l01 * f5, Plo3 = l01 * g5;
        const float Plo4 = l10 * f5, Plo5 = l10 * g5, Plo6 = l11 * f5, Plo7 = l11 * g5;

        // A = SW[ahi x alo]. Rows 8-15 of A feed only D rows 8-15, which no lane
        // reads -> address with (m&7), no zeroing, no predicated loads.
        // Lane layout (ISA 16x16x4 f32): lanes 0-15 K slots {k,k+1}; 16-31 {k+2,k+3}.
        const float* swr = &ldsSW0[n * 64 + (m & 7) * 8];
        const v2f A0 = *(const v2f*)(swr + ks);        // alo base 0
        const v2f A1 = *(const v2f*)(swr + 4 + ks);    // alo base 4
        v2f B0, B1;
        B0.x = hi ? Plo2 : Plo0;  B0.y = hi ? Plo3 : Plo1;
        B1.x = hi ? Plo6 : Plo4;  B1.y = hi ? Plo7 : Plo5;

        v8f acc = {};
        acc = __builtin_amdgcn_wmma_f32_16x16x4_f32(false, A0, false, B0,
                                                    (short)0, acc, false, false);
        acc = __builtin_amdgcn_wmma_f32_16x16x4_f32(false, A1, false, B1,
                                                    (short)0, acc, false, false);

        // lanes 0-15: acc[r] = M[ahi=r] for this lane's pixel; finish h = P_hi . M
        const float h = Phi0 * acc[0] + Phi1 * acc[1] + Phi2 * acc[2] + Phi3 * acc[3]
                      + Phi4 * acc[4] + Phi5 * acc[5] + Phi6 * acc[6] + Phi7 * acc[7];
        if (!hi) ldsH0[n * PIX + p] = h;
    }
    __syncthreads();

    // ---- layer 1: 6 nodes, exact sequential contraction; waves 0-7 fully active ----
    if (tid < PIX) {
        const int pp = tid;
        float g[6];
        for (int n = 0; n < N1; ++n) {
#pragma unroll
            for (int i = 0; i < 6; ++i)
                g[i] = ldsH0[idx1[t * N1 * 6 + n * 6 + i] * PIX + pp];
            ldsH1[n * PIX + pp] = lut64(&ldsSW1[n * 64], g);
        }
    }
    __syncthreads();

    // ---- layer 2: single node -> output ----
    if (tid < PIX) {
        const int pp = tid;
        float g[6];
#pragma unroll
        for (int i = 0; i < 6; ++i) g[i] = ldsH1[idx2[t * 6 + i] * PIX + pp];
        const float h2 = lut64(ldsSW2, g);
        out[((size_t)bImg * NCOUT + t) * PIX + pp] = h2;
    }
}

extern "C" void kernel_launch(void* const* d_in, const int* in_sizes, int n_in,
                              void* d_out, int out_size, void* d_ws, size_t ws_size,
                              hipStream_t stream) {
    (void)in_sizes; (void)n_in; (void)out_size; (void)d_ws; (void)ws_size;
    const float* x      = (const float*)d_in[0];
    const int*   idx0   = (const int*)  d_in[1];
    const float* table0 = (const float*)d_in[2];
    const int*   idx1   = (const int*)  d_in[3];
    const float* table1 = (const float*)d_in[4];
    const int*   idx2   = (const int*)  d_in[5];
    const float* table2 = (const float*)d_in[6];
    float*       out    = (float*)d_out;

    dim3 grid(NCOUT, 8);   // (channel, image)
    dim3 block(512);       // 16 waves; wave w = output row, lane&15 = output col
    lutconv_kernel<<<grid, block, 0, stream>>>(x, idx0, table0, idx1, table1,
                                               idx2, table2, out);
}